// SIR_1597727834171
// MI455X (gfx1250) — compile-verified
//
#include <hip/hip_runtime.h>
#include <stdint.h>

// SIR serial recurrence, one wave32:
//  - bitwise fixed-point early exit, checked once per 8-step group
//    (sound: deterministic map => last-pair equality implies fixedness)
//  - double-buffered LDS ring flushed via CDNA5 async LDS->global stores
//    (global_store_async_from_lds_b128, s_wait_asynccnt / s_wait_dscnt)
//  - constant tail filled by a fully parallel coalesced kernel

#define HALF_STEPS 32                 // steps per half-buffer (384 B)
#define RING_STEPS 64                 // two halves
#define GROUP      8                  // steps per fixed-point check

// One SIR step + buffer traj[tcur] into the LDS ring (lanes 0..2).
#define SIR_STEP(tcur)                                                  \
    {                                                                   \
        float inf = b * S * I;                                          \
        float rec = g * I;                                              \
        float nS = S - inf;                                             \
        float nI = I + inf - rec;                                       \
        float nR = R + rec;                                             \
        S = fminf(fmaxf(nS, 0.0f), pop);                                \
        I = fminf(fmaxf(nI, 0.0f), pop);                                \
        R = fminf(fmaxf(nR, 0.0f), pop);                                \
        int slot_ = (tcur) & (RING_STEPS - 1);                          \
        float v_ = (lane == 0) ? S : ((lane == 1) ? I : R);             \
        if (lane < 3) buf[slot_ * 3 + lane] = v_;                       \
    }

__global__ __launch_bounds__(32)
void sir_serial_kernel(const float* __restrict__ x,
                       const float* __restrict__ bw,
                       const float* __restrict__ gw,
                       const int*   __restrict__ steps_p,
                       float*       __restrict__ out,
                       uint32_t*    __restrict__ ws)
{
    __shared__ float buf[RING_STEPS * 3];   // 768 B LDS ring
    const int lane = (int)threadIdx.x;      // 0..31 (wave32)

    float S = x[0], I = x[1], R = x[2];
    const float pop = S + I + R;            // fixed population (clamp hi)
    const float b = bw[0];
    const float g = gw[0];
    const int n = steps_p[0] - 1;           // trajectory length

    // Low 32 bits of the flat address of a __shared__ object == LDS byte addr.
    const uint32_t lds_base = (uint32_t)(uintptr_t)&buf[0];

    int t = 0;
    bool fixedHit = false;

    while (t + GROUP <= n) {
        SIR_STEP(t + 0)
        SIR_STEP(t + 1)
        SIR_STEP(t + 2)
        SIR_STEP(t + 3)
        SIR_STEP(t + 4)
        SIR_STEP(t + 5)
        SIR_STEP(t + 6)
        float pS = S, pI = I, pR = R;       // state after step t+6
        SIR_STEP(t + 7)
        t += GROUP;

        if ((t & (HALF_STEPS - 1)) == 0) {
            // Flush the just-completed half: steps [t-32, t) = 384 contiguous B.
            asm volatile("s_wait_dscnt 0x0" ::: "memory");  // LDS writes landed
            if (lane < 24) {   // 24 lanes x 16 B = 384 B, one async b128 op
                uint32_t ldsa = lds_base
                              + (uint32_t)(((t - HALF_STEPS) & (RING_STEPS - 1)) * 3 * 4)
                              + (uint32_t)(lane * 16);
                uint64_t ga = (uint64_t)(uintptr_t)(out + (size_t)(t - HALF_STEPS) * 3)
                            + (uint64_t)(lane * 16);
                asm volatile("global_store_async_from_lds_b128 %0, %1, off"
                             :: "v"(ga), "v"(ldsa) : "memory");
            }
            // Before refilling the *other* half (flushed 32 steps ago),
            // ensure that older flush finished reading LDS.
            asm volatile("s_wait_asynccnt 0x1" ::: "memory");
        }

        // Fixed-point check once per group: last consecutive pair only.
        if (S == pS && I == pI && R == pR) { fixedHit = true; break; }
    }

    // Scalar remainder (<= GROUP-1 steps; provably cannot cross a flush
    // boundary since main loop exits within 8 steps of n).
    if (!fixedHit) {
        while (t < n) {
            SIR_STEP(t)
            ++t;
        }
    }

    // Tail: unflushed steps [t & ~31, t), at most 31 entries (93 dwords).
    int flushed = t & ~(HALF_STEPS - 1);
    int rem = t - flushed;
    if (rem > 0) {
        int dwords = rem * 3;
        int slot0 = flushed & (RING_STEPS - 1);
#pragma unroll
        for (int k = 0; k < 3; ++k) {
            int j = k * 32 + lane;
            if (j < dwords) out[(size_t)flushed * 3 + j] = buf[slot0 * 3 + j];
        }
    }

    // Hand off fill start + fixed state to the parallel fill kernel.
    if (lane == 0) {
        ws[0] = (uint32_t)t;                 // == n if never converged
        ws[1] = __float_as_uint(S);
        ws[2] = __float_as_uint(I);
        ws[3] = __float_as_uint(R);
    }
    // S_ENDPGM performs an implicit wait-idle (drains ASYNCcnt/STOREcnt).
}

// Parallel, fully-coalesced constant fill of traj[t_fix .. n-1].
__global__ void sir_fill_kernel(float* __restrict__ out,
                                const uint32_t* __restrict__ ws,
                                int n_el)
{
    int idx = (int)(blockIdx.x * blockDim.x + threadIdx.x);
    if (idx >= n_el) return;
    int start_el = (int)ws[0] * 3;
    if (idx < start_el) return;
    int c = idx % 3;
    out[idx] = __uint_as_float(ws[1 + c]);
}

extern "C" void kernel_launch(void* const* d_in, const int* in_sizes, int n_in,
                              void* d_out, int out_size, void* d_ws, size_t ws_size,
                              hipStream_t stream)
{
    (void)in_sizes; (void)n_in; (void)ws_size;
    const float* x     = (const float*)d_in[0];
    const float* bw    = (const float*)d_in[1];
    const float* gw    = (const float*)d_in[2];
    const int*   steps = (const int*)d_in[3];
    float*       out   = (float*)d_out;
    uint32_t*    ws    = (uint32_t*)d_ws;

    sir_serial_kernel<<<1, 32, 0, stream>>>(x, bw, gw, steps, out, ws);

    int blocks = (out_size + 255) / 256;
    if (blocks < 1) blocks = 1;
    sir_fill_kernel<<<blocks, 256, 0, stream>>>(out, ws, out_size);
}